// PredictModel_17772574670885
// MI455X (gfx1250) — compile-verified
//
#include <hip/hip_runtime.h>
#include <stdint.h>

// ---------------- problem constants ----------------
#define B_      8
#define A_      110000
#define C_      80
#define TOPK_   200
#define PRE_    1000
#define NPAD_   1024
#define CONF_T_ 0.05f
#define NMS_T_  0.5f
#define CROP_   512.0f
#define CLS_OFF_ 2.0f
#define HISTN_  65536

// ---------------- classmax tiling ----------------
#define ANCH_        128          // anchors per tile
#define TPB_CM_      128          // threads per block (4 waves of 32)
#define LDS_STRIDE_  336          // 84 dwords per anchor row (80 data + 4 pad) -> ~conflict-free b128 reads
#define TILE_BYTES_  (ANCH_ * LDS_STRIDE_)   // 43008 B ; double buffered = 86016 B LDS
#define CHUNKS_PA_   20           // 320 B per anchor / 16 B
#define CM_GRIDX_    128          // tiles strided across 128 blocks per image

// ---------------- CDNA5 async global->LDS helpers ----------------
typedef int v4i_ __attribute__((vector_size(16)));                  // matches builtin's param type
typedef __attribute__((address_space(1))) v4i_* as1_v4ip;           // global
typedef __attribute__((address_space(3))) v4i_* as3_v4ip;           // LDS

__device__ __forceinline__ void async_copy16(const void* g, unsigned lds_off) {
#if __has_builtin(__builtin_amdgcn_global_load_async_to_lds_b128)
  __builtin_amdgcn_global_load_async_to_lds_b128(
      (as1_v4ip)(uintptr_t)g, (as3_v4ip)lds_off, 0, 0);
#else
  asm volatile("global_load_async_to_lds_b128 %0, %1, off"
               :: "v"(lds_off), "v"(g) : "memory");
#endif
}

template <int N>
__device__ __forceinline__ void wait_asynccnt() {
#if __has_builtin(__builtin_amdgcn_s_wait_asynccnt)
  __builtin_amdgcn_s_wait_asynccnt((short)N);
#else
  asm volatile("s_wait_asynccnt %0" :: "i"(N) : "memory");
#endif
}

// ---------------- stage 1: class max/argmax via async-LDS staging ----------------
// (first kernel in the TU so the disasm snippet shows the async prefetch burst)
__device__ __forceinline__ void prefetch_tile(const char* gb, unsigned lbase,
                                              int nchunks, int t) {
#pragma unroll
  for (int k = 0; k < CHUNKS_PA_; ++k) {
    int g = k * TPB_CM_ + t;
    g = min(g, nchunks - 1);                       // clamp: keeps ASYNCcnt increments uniform (20/wave)
    unsigned a = (unsigned)g / 20u;
    unsigned c = (unsigned)g - a * 20u;
    async_copy16(gb + (size_t)g * 16, lbase + a * LDS_STRIDE_ + c * 16);
  }
}

__global__ __launch_bounds__(TPB_CM_) void k_classmax(
    const float* __restrict__ conf, float* __restrict__ smax,
    int* __restrict__ cls, unsigned* __restrict__ hist1) {
  __shared__ __align__(16) char smem[2 * TILE_BYTES_];
  const int t = threadIdx.x;
  const int b = blockIdx.y;
  const int ntiles = (A_ + ANCH_ - 1) / ANCH_;

  const unsigned lb0 = (unsigned)(uintptr_t)(void*)&smem[0];
  const unsigned lb1 = (unsigned)(uintptr_t)(void*)&smem[TILE_BYTES_];

  int tile = blockIdx.x;
  if (tile < ntiles) {
    int nrem = min(ANCH_, A_ - tile * ANCH_);
    const char* gb = (const char*)(conf + ((size_t)b * A_ + (size_t)tile * ANCH_) * C_);
    prefetch_tile(gb, lb0, nrem * CHUNKS_PA_, t);
  }
  int it = 0;
  for (; tile < ntiles; tile += gridDim.x, ++it) {
    int nxt = tile + gridDim.x;
    if (nxt < ntiles) {
      int nrem2 = min(ANCH_, A_ - nxt * ANCH_);
      const char* gb2 = (const char*)(conf + ((size_t)b * A_ + (size_t)nxt * ANCH_) * C_);
      prefetch_tile(gb2, (it & 1) ? lb0 : lb1, nrem2 * CHUNKS_PA_, t);
      wait_asynccnt<CHUNKS_PA_>();                 // current tile landed; next tile in flight
    } else {
      wait_asynccnt<0>();
    }
    __syncthreads();
    const char* cb = smem + ((it & 1) ? TILE_BYTES_ : 0);
    int a0 = tile * ANCH_;
    int nrem = min(ANCH_, A_ - a0);
    if (t < nrem) {
      const float4* rp = (const float4*)(cb + t * LDS_STRIDE_);
      float best = -1.0f; int bi = 0;
#pragma unroll
      for (int j = 0; j < 20; ++j) {
        float4 v = rp[j]; int base = j * 4;
        if (v.x > best) { best = v.x; bi = base; }
        if (v.y > best) { best = v.y; bi = base + 1; }
        if (v.z > best) { best = v.z; bi = base + 2; }
        if (v.w > best) { best = v.w; bi = base + 3; }
      }
      size_t gi = (size_t)b * A_ + a0 + t;
      float m = (best > CONF_T_) ? best : -1.0f;
      smax[gi] = m; cls[gi] = bi;
      if (m > 0.f) {
        unsigned bits = __float_as_uint(m);
        atomicAdd(&hist1[(size_t)b * HISTN_ + (bits >> 16)], 1u);
      }
    }
    __syncthreads();
  }
}

// ---------------- init ----------------
__global__ void k_init(unsigned* hist1, unsigned* hist2, unsigned* info,
                       float* candS, int* candA) {
  size_t i = (size_t)blockIdx.x * blockDim.x + threadIdx.x;
  if (i < (size_t)B_ * HISTN_) { hist1[i] = 0u; hist2[i] = 0u; }
  if (i < (size_t)B_ * 32)     info[i] = 0u;
  if (i < (size_t)B_ * PRE_)   { candS[i] = -1.0f; candA[i] = 0; }
}

// ---------------- stage 2: two-level radix select of the top-PRE cutoff ----------------
__global__ __launch_bounds__(1024) void k_cut1(const unsigned* __restrict__ hist,
                                               unsigned* __restrict__ info) {
  const int b = blockIdx.x, t = threadIdx.x;
  __shared__ unsigned part[1024];
  const unsigned* h = hist + (size_t)b * HISTN_;
  unsigned s = 0; int hi = HISTN_ - 1 - t * 64;
  for (int j = 0; j < 64; ++j) s += h[hi - j];
  part[t] = s; __syncthreads();
  if (t == 0) {
    unsigned* inf = info + b * 32;
    unsigned cum = 0; int pick = -1;
    for (int k = 0; k < 1024; ++k) { if (cum + part[k] >= PRE_) { pick = k; break; } cum += part[k]; }
    if (pick < 0) { inf[0] = 0xFFFFFFFFu; inf[1] = cum; }     // fewer than PRE valid -> take all
    else {
      int top = HISTN_ - 1 - pick * 64; unsigned c2 = cum; int bin = top;
      for (int j = 0; j < 64; ++j) { unsigned hv = h[top - j]; if (c2 + hv >= PRE_) { bin = top - j; break; } c2 += hv; }
      inf[0] = (unsigned)bin; inf[1] = c2;                    // c2 = count strictly above bin
    }
  }
}

__global__ void k_hist2(const float* __restrict__ smax, const unsigned* __restrict__ info,
                        unsigned* __restrict__ hist2) {
  size_t i = (size_t)blockIdx.x * blockDim.x + threadIdx.x;
  if (i >= (size_t)B_ * A_) return;
  int b = (int)(i / A_);
  float s = smax[i]; if (s <= 0.f) return;
  unsigned bin1 = info[b * 32 + 0]; if (bin1 == 0xFFFFFFFFu) return;
  unsigned bits = __float_as_uint(s);
  if ((bits >> 16) == bin1) atomicAdd(&hist2[(size_t)b * HISTN_ + (bits & 0xFFFFu)], 1u);
}

__global__ __launch_bounds__(1024) void k_cut2(const unsigned* __restrict__ hist2,
                                               unsigned* __restrict__ info) {
  const int b = blockIdx.x, t = threadIdx.x;
  __shared__ unsigned part[1024];
  unsigned* inf = info + b * 32;
  const unsigned* h = hist2 + (size_t)b * HISTN_;
  unsigned s = 0; int hi = HISTN_ - 1 - t * 64;
  for (int j = 0; j < 64; ++j) s += h[hi - j];
  part[t] = s; __syncthreads();
  if (t == 0) {
    if (inf[0] == 0xFFFFFFFFu) { inf[2] = 0u; inf[3] = 0u; inf[4] = 0u; }
    else {
      unsigned target = PRE_ - inf[1];
      unsigned cum = 0; int pick = 0;
      for (int k = 0; k < 1024; ++k) { if (cum + part[k] >= target) { pick = k; break; } cum += part[k]; }
      int top = HISTN_ - 1 - pick * 64; unsigned c2 = cum; int low = top;
      for (int j = 0; j < 64; ++j) { unsigned hv = h[top - j]; if (c2 + hv >= target) { low = top - j; break; } c2 += hv; }
      inf[2] = (inf[0] << 16) | (unsigned)low;  // full 32-bit cutoff key
      inf[3] = inf[1] + c2;                     // n_gt: strictly greater than cutoff
      inf[4] = PRE_ - inf[3];                   // ties admitted at cutoff
    }
  }
}

__global__ void k_compact(const float* __restrict__ smax, unsigned* __restrict__ info,
                          float* __restrict__ candS, int* __restrict__ candA) {
  size_t i = (size_t)blockIdx.x * blockDim.x + threadIdx.x;
  if (i >= (size_t)B_ * A_) return;
  int b = (int)(i / A_); int a = (int)(i - (size_t)b * A_);
  float s = smax[i]; if (s <= 0.f) return;
  unsigned* inf = info + b * 32;
  unsigned bits = __float_as_uint(s), ck = inf[2];
  if (bits > ck) {
    unsigned p = atomicAdd(&inf[5], 1u);
    if (p < PRE_) { candS[b * PRE_ + p] = s; candA[b * PRE_ + p] = a; }
  } else if (bits == ck) {
    unsigned need = inf[4];
    unsigned j = atomicAdd(&inf[6], 1u);
    if (j < need) { unsigned p = inf[3] + j; candS[b * PRE_ + p] = s; candA[b * PRE_ + p] = a; }
  }
}

// ---------------- stage 3: decode the 1000 selected boxes per image ----------------
__global__ void k_decode(const float* __restrict__ reg, const float* __restrict__ anch,
                         const int* __restrict__ cls, const int* __restrict__ candA,
                         float* __restrict__ candBox, float* __restrict__ candBo,
                         int* __restrict__ candC) {
  int i = blockIdx.x * blockDim.x + threadIdx.x;
  if (i >= B_ * PRE_) return;
  int b = i / PRE_;
  int a = candA[i];
  float a0 = anch[a * 4 + 0], a1 = anch[a * 4 + 1], a2 = anch[a * 4 + 2], a3 = anch[a * 4 + 3];
  float ya = 0.5f * (a0 + a2), xa = 0.5f * (a1 + a3);
  float ha = a2 - a0, wa = a3 - a1;
  const float* r = reg + ((size_t)b * A_ + a) * 4;
  float yc = r[0] * ha + ya, xc = r[1] * wa + xa;
  float hh = expf(r[2]) * ha, ww = expf(r[3]) * wa;
  float v[4] = { xc - 0.5f * ww, yc - 0.5f * hh, xc + 0.5f * ww, yc + 0.5f * hh };
  int c = cls[(size_t)b * A_ + a];
  float off = (float)c * CLS_OFF_;
#pragma unroll
  for (int k = 0; k < 4; ++k) {
    float cv = fminf(fmaxf(v[k], 0.f), CROP_) * (1.0f / CROP_);
    candBox[(size_t)i * 4 + k] = cv;
    candBo[(size_t)i * 4 + k]  = cv + off;
  }
  candC[i] = c;
}

// ---------------- bitonic sort (descending, N=1024) ----------------
__device__ __forceinline__ void bitonic_desc(float* ks, int* ps, int t) {
  for (int k = 2; k <= NPAD_; k <<= 1) {
    for (int j = k >> 1; j > 0; j >>= 1) {
      int x = t ^ j;
      if (x > t) {
        bool up = ((t & k) == 0);
        float a = ks[t], b = ks[x];
        bool sw = up ? (a < b) : (a > b);
        if (sw) { ks[t] = b; ks[x] = a; int tp = ps[t]; ps[t] = ps[x]; ps[x] = tp; }
      }
      __syncthreads();
    }
  }
}

__global__ __launch_bounds__(1024) void k_sortcand(const float* __restrict__ candS,
                                                   int* __restrict__ perm,
                                                   float* __restrict__ sscore) {
  int b = blockIdx.x, t = threadIdx.x;
  __shared__ float ks[NPAD_]; __shared__ int ps[NPAD_];
  ks[t] = (t < PRE_) ? candS[b * PRE_ + t] : -1e30f;
  ps[t] = (t < PRE_) ? t : 0;
  __syncthreads();
  bitonic_desc(ks, ps, t);
  perm[b * NPAD_ + t] = ps[t];
  sscore[b * NPAD_ + t] = ks[t];
}

// ---------------- stage 4: suppression bitmask + sequential greedy reduce ----------------
__global__ __launch_bounds__(64) void k_supmat(const float* __restrict__ candBo,
                                               const int* __restrict__ perm,
                                               unsigned long long* __restrict__ sup) {
  int b = blockIdx.z, rt = blockIdx.y, ct = blockIdx.x, t = threadIdx.x;
  __shared__ float cb[64][4]; __shared__ float ca[64];
  int colr = ct * 64 + t;
  int cslot = perm[b * NPAD_ + colr];
  const float* pb = candBo + ((size_t)b * PRE_ + cslot) * 4;
  cb[t][0] = pb[0]; cb[t][1] = pb[1]; cb[t][2] = pb[2]; cb[t][3] = pb[3];
  ca[t] = (pb[2] - pb[0]) * (pb[3] - pb[1]);
  __syncthreads();
  int row = rt * 64 + t;
  int rslot = perm[b * NPAD_ + row];
  const float* rb = candBo + ((size_t)b * PRE_ + rslot) * 4;
  float r0 = rb[0], r1 = rb[1], r2 = rb[2], r3 = rb[3];
  float rar = (r2 - r0) * (r3 - r1);
  unsigned long long bits = 0ull;
  for (int j = 0; j < 64; ++j) {
    int c = ct * 64 + j;
    if (c <= row) continue;
    float lx = fmaxf(r0, cb[j][0]), ly = fmaxf(r1, cb[j][1]);
    float hx = fminf(r2, cb[j][2]), hy = fminf(r3, cb[j][3]);
    float iw = fmaxf(hx - lx, 0.f), ih = fmaxf(hy - ly, 0.f);
    float inter = iw * ih;
    float iou = inter / (rar + ca[j] - inter + 1e-8f);
    if (iou > NMS_T_) bits |= (1ull << j);
  }
  sup[((size_t)b * NPAD_ + row) * 16 + ct] = bits;
}

__global__ void k_nmsreduce(const unsigned long long* __restrict__ sup,
                            const float* __restrict__ sscore,
                            unsigned long long* __restrict__ keepm) {
  int b = blockIdx.x, lane = threadIdx.x;
  unsigned long long validw = 0ull, remv = 0ull;
  if (lane < 16) {
    for (int k = 0; k < 64; ++k) {
      int r = lane * 64 + k;
      if (r < PRE_ && sscore[b * NPAD_ + r] > 0.f) validw |= (1ull << k);
    }
  }
  for (int i = 0; i < PRE_; ++i) {
    int w = i >> 6, bt = i & 63;
    unsigned long long rw = __shfl(remv, w);
    unsigned long long vw = __shfl(validw, w);
    bool alive = !((rw >> bt) & 1ull) && ((vw >> bt) & 1ull);
    if (alive && lane < 16) remv |= sup[((size_t)b * NPAD_ + i) * 16 + lane];
  }
  if (lane < 16) keepm[b * 16 + lane] = validw & ~remv;
}

// ---------------- stage 5: per-class top-200 output ----------------
__global__ __launch_bounds__(1024) void k_perclass(const float* __restrict__ sscore,
    const int* __restrict__ perm, const int* __restrict__ candC,
    const float* __restrict__ candBox, const unsigned long long* __restrict__ keepm,
    float* __restrict__ out) {
  int ci = blockIdx.x, b = blockIdx.y, t = threadIdx.x;
  __shared__ float ks[NPAD_]; __shared__ int ps[NPAD_];
  float s = -1e30f;
  if (t < PRE_) {
    float sc = sscore[b * NPAD_ + t];
    bool kept = (keepm[b * 16 + (t >> 6)] >> (t & 63)) & 1ull;
    if (kept && sc > 0.f) {
      int slot = perm[b * NPAD_ + t];
      if (candC[b * PRE_ + slot] == ci) s = sc;
    }
  }
  ks[t] = s; ps[t] = t; __syncthreads();
  bitonic_desc(ks, ps, t);
  if (t < TOPK_) {
    float* o = out + (((size_t)(b * C_ + ci)) * TOPK_ + t) * 5;
    float sv = ks[t];
    if (sv > 0.f) {
      int slot = perm[b * NPAD_ + ps[t]];
      const float* bx = candBox + ((size_t)b * PRE_ + slot) * 4;
      o[0] = bx[0]; o[1] = bx[1]; o[2] = bx[2]; o[3] = bx[3]; o[4] = sv;
    } else {
      o[0] = 0.f; o[1] = 0.f; o[2] = 0.f; o[3] = 0.f; o[4] = 0.f;
    }
  }
}

// ---------------- host launcher ----------------
extern "C" void kernel_launch(void* const* d_in, const int* in_sizes, int n_in,
                              void* d_out, int out_size, void* d_ws, size_t ws_size,
                              hipStream_t stream) {
  (void)in_sizes; (void)n_in; (void)out_size; (void)ws_size;
  const float* conf = (const float*)d_in[0];   // (B, A, C)
  const float* reg  = (const float*)d_in[1];   // (B, A, 4)
  const float* anch = (const float*)d_in[2];   // (A, 4)
  float* out = (float*)d_out;                  // (B, C, TOPK, 5)

  char* ws = (char*)d_ws;
  size_t off = 0;
  auto take = [&](size_t n) { size_t r = off; off += (n + 255) & ~(size_t)255; return r; };
  float*    smax  = (float*)(ws + take((size_t)B_ * A_ * 4));
  int*      cls   = (int*)(ws + take((size_t)B_ * A_ * 4));
  unsigned* hist1 = (unsigned*)(ws + take((size_t)B_ * HISTN_ * 4));
  unsigned* hist2 = (unsigned*)(ws + take((size_t)B_ * HISTN_ * 4));
  unsigned* info  = (unsigned*)(ws + take((size_t)B_ * 32 * 4));
  float*    candS = (float*)(ws + take((size_t)B_ * PRE_ * 4));
  int*      candA = (int*)(ws + take((size_t)B_ * PRE_ * 4));
  float*    candBox = (float*)(ws + take((size_t)B_ * PRE_ * 16));
  float*    candBo  = (float*)(ws + take((size_t)B_ * PRE_ * 16));
  int*      candC   = (int*)(ws + take((size_t)B_ * PRE_ * 4));
  int*      perm    = (int*)(ws + take((size_t)B_ * NPAD_ * 4));
  float*    sscore  = (float*)(ws + take((size_t)B_ * NPAD_ * 4));
  unsigned long long* sup   = (unsigned long long*)(ws + take((size_t)B_ * NPAD_ * 16 * 8));
  unsigned long long* keepm = (unsigned long long*)(ws + take((size_t)B_ * 16 * 8));

  int initN = B_ * HISTN_;
  k_init<<<(initN + 255) / 256, 256, 0, stream>>>(hist1, hist2, info, candS, candA);

  k_classmax<<<dim3(CM_GRIDX_, B_), TPB_CM_, 0, stream>>>(conf, smax, cls, hist1);

  k_cut1<<<B_, 1024, 0, stream>>>(hist1, info);

  int nBA = B_ * A_;
  k_hist2<<<(nBA + 255) / 256, 256, 0, stream>>>(smax, info, hist2);
  k_cut2<<<B_, 1024, 0, stream>>>(hist2, info);
  k_compact<<<(nBA + 255) / 256, 256, 0, stream>>>(smax, info, candS, candA);

  k_decode<<<(B_ * PRE_ + 127) / 128, 128, 0, stream>>>(reg, anch, cls, candA,
                                                        candBox, candBo, candC);
  k_sortcand<<<B_, 1024, 0, stream>>>(candS, perm, sscore);

  k_supmat<<<dim3(16, 16, B_), 64, 0, stream>>>(candBo, perm, sup);
  k_nmsreduce<<<B_, 32, 0, stream>>>(sup, sscore, keepm);

  k_perclass<<<dim3(C_, B_), 1024, 0, stream>>>(sscore, perm, candC, candBox, keepm, out);
}